// BasicConv_78915729097031
// MI455X (gfx1250) — compile-verified
//
#include <hip/hip_runtime.h>
#include <math.h>

typedef __attribute__((ext_vector_type(2))) float v2f;
typedef __attribute__((ext_vector_type(8))) float v8f;

#define NNODES 8192
#define NEDGES 65536
#define FEAT   128   // 4*MUL
#define EMB    64
#define HID    128
#define WCOLS  4096

#define INV3   0.57735026918962576f   // 1/sqrt(3)
#define GAIN   1.679f
#define W1SC   0.125f                 // 1/sqrt(64)
#define W2SC   0.08838834764831845f   // 1/sqrt(128)
#define NORM0  0.125f                 // sqrt(1/64)
#define NORM1  0.21650635094610965f   // sqrt(3/64)
#define INVNB  0.35355339059327373f   // 1/sqrt(8)

__device__ __forceinline__ v8f wmma4(v2f a, v2f b, v8f c) {
  // D = A(16x4 f32) * B(4x16 f32) + C(16x16 f32)
  return __builtin_amdgcn_wmma_f32_16x16x4_f32(false, a, false, b, (short)0, c,
                                               false, false);
}

// Pre-pass: W2T[col*128 + c] = W2[c*4096 + col] / sqrt(128)
__global__ void transpose_w2(const float* __restrict__ w2,
                             float* __restrict__ w2t) {
  int idx = blockIdx.x * 256 + threadIdx.x;   // 0 .. 524287
  int col = idx >> 7;                         // 0..4095
  int c   = idx & 127;                        // 0..127
  w2t[idx] = w2[c * WCOLS + col] * W2SC;
}

__global__ __launch_bounds__(64) void fused_conv(
    const float* __restrict__ x, const int* __restrict__ eidx,
    const float* __restrict__ attr, const float* __restrict__ emb,
    const float* __restrict__ w1, const float* __restrict__ w2t,
    float* __restrict__ out) {
  // Per-wave LDS region: h[16][128] (2048) | cA[32][16] | cB[32][16]
  //                      | cC[32][3][16] | cD[32][3][16]  = 6144 floats
  __shared__ __align__(16) float sm[2 * 6144];
  __shared__ int jsm[2][16];

  const int wave = threadIdx.x >> 5;
  const int lane = threadIdx.x & 31;
  const int lo = lane & 15;     // N / M-row lane index
  const int hh = lane >> 4;     // lane half (K / M-block select)

  float* H  = sm + wave * 6144;
  float* cA = H + 2048;
  float* cB = cA + 512;
  float* cC = cB + 512;
  float* cD = cC + 1536;

  const int ebase = (blockIdx.x * 2 + wave) * 16;

  // ---------------- GEMM1: h = GAIN * silu((emb*W1SC) @ W1), 16x128 tile ----
  v2f a1[16];
  {
    const float* er = emb + (size_t)(ebase + lo) * EMB + 2 * hh;
#pragma unroll
    for (int k = 0; k < 16; ++k) {
      v2f t = *(const v2f*)(er + 4 * k);
      a1[k] = t * W1SC;   // fold 1/sqrt(64) into A
    }
  }
#pragma unroll 1
  for (int nc = 0; nc < 8; ++nc) {
    v8f acc = {};
#pragma unroll
    for (int k = 0; k < 16; ++k) {
      const float* bp = w1 + (size_t)(4 * k + 2 * hh) * HID + nc * 16 + lo;
      v2f b;
      b.x = bp[0];
      b.y = bp[HID];
      acc = wmma4(a1[k], b, acc);
    }
#pragma unroll
    for (int r = 0; r < 8; ++r) {
      float v = acc[r];
      float s = GAIN * v / (1.0f + __expf(-v));     // gain * silu
      H[(r + 8 * hh) * HID + nc * 16 + lo] = s;
    }
  }

  // ---------------- Per-edge TP coefficients -> LDS [u][edge] --------------
  {
    const int u = lane;                              // u = 0..31
#pragma unroll 1
    for (int e = 0; e < 16; ++e) {
      int ie = eidx[ebase + e];
      const float* at = attr + (size_t)(ebase + e) * 4;
      float s0 = at[0], s1 = at[1], s2 = at[2], s3 = at[3];
      const float* xr = x + (size_t)ie * FEAT;
      float x0 = xr[u];
      float y0 = xr[32 + u * 3 + 0];
      float y1 = xr[32 + u * 3 + 1];
      float y2 = xr[32 + u * 3 + 2];
      cA[u * 16 + e] = x0 * s0;                          // path Wa
      cB[u * 16 + e] = (y0 * s1 + y1 * s2 + y2 * s3) * INV3;  // path Wb
      float xi = x0 * INV3;
      cC[(u * 3 + 0) * 16 + e] = xi * s1;                // path Wc (per k)
      cC[(u * 3 + 1) * 16 + e] = xi * s2;
      cC[(u * 3 + 2) * 16 + e] = xi * s3;
      float sb = s0 * INV3;
      cD[(u * 3 + 0) * 16 + e] = y0 * sb;                // path Wd (per k)
      cD[(u * 3 + 1) * 16 + e] = y1 * sb;
      cD[(u * 3 + 2) * 16 + e] = y2 * sb;
      if (lane == 0) jsm[wave][e] = eidx[NEDGES + ebase + e];
    }
  }
  __syncthreads();

  // ---------------- Preload GEMM2 A-fragments (h rows) ---------------------
  v2f aF[32];
#pragma unroll
  for (int k = 0; k < 32; ++k)
    aF[k] = *(const v2f*)(H + lo * HID + 4 * k + 2 * hh);

  v8f o0[2] = {};          // out0, two 16-col halves
  v8f o1[3][2] = {};       // out1[k], two 16-col halves

  // One (p,u,half) chunk: 16 edges x 16 cols over K=128, 32 wmma
  auto gemmChunk = [&](int col0) -> v8f {
    const float* bp = w2t + (size_t)(col0 + lo) * HID + 2 * hh;
    __builtin_prefetch(bp + 32 * HID, 0, 0);   // next chunk's B block
    v8f acc = {};
#pragma unroll
    for (int k = 0; k < 32; ++k) {
      v2f b = *(const v2f*)(bp + 4 * k);
      acc = wmma4(aF[k], b, acc);
    }
    return acc;
  };

  // ---- p = 0 (Wa -> out0) and p = 1 (Wb -> out0) --------------------------
#pragma unroll 1
  for (int uu = 0; uu < 32; ++uu) {
    const float4 al = *(const float4*)(cA + uu * 16 + hh * 8);
    const float4 ah = *(const float4*)(cA + uu * 16 + hh * 8 + 4);
    float cf[8] = {al.x, al.y, al.z, al.w, ah.x, ah.y, ah.z, ah.w};
#pragma unroll
    for (int nh = 0; nh < 2; ++nh) {
      v8f d = gemmChunk(uu * 32 + nh * 16);
#pragma unroll
      for (int r = 0; r < 8; ++r) o0[nh][r] += cf[r] * d[r];
    }
  }
#pragma unroll 1
  for (int uu = 0; uu < 32; ++uu) {
    const float4 al = *(const float4*)(cB + uu * 16 + hh * 8);
    const float4 ah = *(const float4*)(cB + uu * 16 + hh * 8 + 4);
    float cf[8] = {al.x, al.y, al.z, al.w, ah.x, ah.y, ah.z, ah.w};
#pragma unroll
    for (int nh = 0; nh < 2; ++nh) {
      v8f d = gemmChunk(1024 + uu * 32 + nh * 16);
#pragma unroll
      for (int r = 0; r < 8; ++r) o0[nh][r] += cf[r] * d[r];
    }
  }
  // ---- p = 2 (Wc -> out1[k]) ----------------------------------------------
#pragma unroll 1
  for (int uu = 0; uu < 32; ++uu) {
#pragma unroll
    for (int nh = 0; nh < 2; ++nh) {
      v8f d = gemmChunk(2048 + uu * 32 + nh * 16);
#pragma unroll
      for (int kk = 0; kk < 3; ++kk) {
        const float4 al = *(const float4*)(cC + (uu * 3 + kk) * 16 + hh * 8);
        const float4 ah = *(const float4*)(cC + (uu * 3 + kk) * 16 + hh * 8 + 4);
        float cf[8] = {al.x, al.y, al.z, al.w, ah.x, ah.y, ah.z, ah.w};
#pragma unroll
        for (int r = 0; r < 8; ++r) o1[kk][nh][r] += cf[r] * d[r];
      }
    }
  }
  // ---- p = 3 (Wd -> out1[k]) ----------------------------------------------
#pragma unroll 1
  for (int uu = 0; uu < 32; ++uu) {
#pragma unroll
    for (int nh = 0; nh < 2; ++nh) {
      v8f d = gemmChunk(3072 + uu * 32 + nh * 16);
#pragma unroll
      for (int kk = 0; kk < 3; ++kk) {
        const float4 al = *(const float4*)(cD + (uu * 3 + kk) * 16 + hh * 8);
        const float4 ah = *(const float4*)(cD + (uu * 3 + kk) * 16 + hh * 8 + 4);
        float cf[8] = {al.x, al.y, al.z, al.w, ah.x, ah.y, ah.z, ah.w};
#pragma unroll
        for (int r = 0; r < 8; ++r) o1[kk][nh][r] += cf[r] * d[r];
      }
    }
  }

  // ---------------- Scale + scatter-add into destination nodes -------------
  const float sc0 = NORM0 * INVNB;
  const float sc1 = NORM1 * INVNB;
#pragma unroll
  for (int r = 0; r < 8; ++r) {
    int e = r + 8 * hh;
    int node = jsm[wave][e];
    float* orow = out + (size_t)node * FEAT;
#pragma unroll
    for (int nh = 0; nh < 2; ++nh) {
      int c = nh * 16 + lo;   // output channel w within 0..31
      unsafeAtomicAdd(orow + c, o0[nh][r] * sc0);
#pragma unroll
      for (int kk = 0; kk < 3; ++kk)
        unsafeAtomicAdd(orow + 32 + c * 3 + kk, o1[kk][nh][r] * sc1);
    }
  }
}

extern "C" void kernel_launch(void* const* d_in, const int* in_sizes, int n_in,
                              void* d_out, int out_size, void* d_ws,
                              size_t ws_size, hipStream_t stream) {
  const float* x    = (const float*)d_in[0];
  const int*   eidx = (const int*)d_in[1];
  const float* attr = (const float*)d_in[2];
  const float* emb  = (const float*)d_in[3];
  const float* w1   = (const float*)d_in[4];
  const float* w2   = (const float*)d_in[5];
  float* out = (float*)d_out;
  float* w2t = (float*)d_ws;   // 128*4096 floats = 2 MB scratch

  hipMemsetAsync(d_out, 0, (size_t)out_size * sizeof(float), stream);
  transpose_w2<<<(HID * WCOLS) / 256, 256, 0, stream>>>(w2, w2t);
  fused_conv<<<NEDGES / 32, 64, 0, stream>>>(x, eidx, attr, emb, w1, w2t, out);
}